// SpatialOversizeConv2d_8693013807281
// MI455X (gfx1250) — compile-verified
//
#include <hip/hip_runtime.h>

typedef __attribute__((ext_vector_type(2))) float v2f;
typedef __attribute__((ext_vector_type(8))) float v8f;

#define DIMC 256
#define NB   10
#define HW   128
#define KTAP 255
#define KPAD 272   // GEMM K dim: 15 (tile M span) + 255 taps, padded to mult of 4
#define WLEN 288   // padded weights: [16 zeros][255 taps][17 zeros]
#define VST  40    // vertical lds row stride (floats): 2*40 mod 64 = 16 -> no bank conflict
#define HST  388   // horizontal lds row stride: 388 mod 64 = 4 -> conflict-free column reads

#define WMMA_F32X4(A, B, C) \
  __builtin_amdgcn_wmma_f32_16x16x4_f32(false, (A), false, (B), (short)0, (C), false, false)

// ---------------- vertical depthwise conv (255,1), pad (127,0) ----------------
// block (128 thr = 4 waves) = (nc, 32-col slab). Wave w: h-tiles {32w, 32w+16} x
// w-tiles {0,16}. Weight-Toeplitz A-frag shared by all 4 tiles: 10 DS / 4 WMMA.
// Tile GEMM: Out[m,n] = sum_j T[m,j] * Xpad[h0+j-127, w0+n],  T[m,j]=W[j-m]
__global__ __launch_bounds__(128)
void dwconv_v_wmma(const float* __restrict__ x,
                   const float* __restrict__ wh,
                   const float* __restrict__ bh,
                   float* __restrict__ y) {
  __shared__ __align__(16) float ldsX[384 * VST];  // rows: h+127, h in [-127,256]
  __shared__ __align__(16) float wpad[WLEN];

  const int tid = threadIdx.x;
  const int nc  = blockIdx.x >> 2;          // 0..2559
  const int c   = nc % DIMC;
  const int wq  = (blockIdx.x & 3) << 5;    // 0,32,64,96

  const float4 z4 = make_float4(0.f, 0.f, 0.f, 0.f);
  for (int i = tid; i < (384 * VST) / 4; i += 128) ((float4*)ldsX)[i] = z4;
  for (int i = tid; i < WLEN; i += 128) {
    int k = i - 16;
    wpad[i] = (k >= 0 && k < KTAP) ? wh[c * KTAP + k] : 0.0f;
  }
  __syncthreads();
  {
    const float* src = x + (size_t)nc * HW * HW + wq;
    int col = tid & 31, rb = tid >> 5;      // 32 lanes -> 128B bursts
    for (int i = 0; i < 32; ++i) {
      int h = rb + 4 * i;
      ldsX[(h + 127) * VST + col] = src[h * HW + col];
    }
  }
  __syncthreads();

  const int lane = tid & 31;
  const int hp   = (tid >> 5) << 5;         // wave -> 0,32,64,96
  const int half = lane >> 4;
  const int ln   = lane & 15;
  const int kkb  = half << 1;               // K-subrow base: 0 | 2

  const float bias = bh[c];
  v8f acc00, acc01, acc10, acc11;           // [hIdx][wIdx]
  for (int r = 0; r < 8; ++r) { acc00[r] = bias; acc01[r] = bias; acc10[r] = bias; acc11[r] = bias; }

  for (int j0 = 0; j0 < KPAD; j0 += 4) {
    int wi = 16 + j0 + kkb - ln;            // T[m, j0+kkb] = W[j0+kkb-m]
    v2f a = {wpad[wi], wpad[wi + 1]};
    int r0 = hp + j0 + kkb;                 // lds row = (h0-127+j) + 127
    const float* p0 = &ldsX[r0 * VST + ln];
    const float* p1 = p0 + 16 * VST;
    v2f b00 = {p0[0],  p0[VST]};
    v2f b01 = {p0[16], p0[VST + 16]};
    v2f b10 = {p1[0],  p1[VST]};
    v2f b11 = {p1[16], p1[VST + 16]};
    acc00 = WMMA_F32X4(a, b00, acc00);
    acc01 = WMMA_F32X4(a, b01, acc01);
    acc10 = WMMA_F32X4(a, b10, acc10);
    acc11 = WMMA_F32X4(a, b11, acc11);
  }

  float* dst = y + (size_t)nc * HW * HW;
  const int rb0 = hp + (half << 3);         // C/D: vgpr r -> rows r, r+8
  const int rb1 = rb0 + 16;
  const int c0  = wq + ln;
  for (int r = 0; r < 8; ++r) {
    dst[(rb0 + r) * HW + c0]      = acc00[r];
    dst[(rb0 + r) * HW + c0 + 16] = acc01[r];
    dst[(rb1 + r) * HW + c0]      = acc10[r];
    dst[(rb1 + r) * HW + c0 + 16] = acc11[r];
  }
}

// ---------------- horizontal depthwise conv (1,255), pad (0,127) --------------
// block (256 thr = 8 waves) = (nc, 32-row slab). Wave w: w-tile 16w, h-tiles {0,16};
// weight B-frag shared across the 2 tiles: 6 DS / 2 WMMA.
// Tile GEMM: Out[m,n] = sum_j Ypad[h0+m, w0+j-127] * T[j,n],  T[j,n]=W[j-n]
__global__ __launch_bounds__(256)
void dwconv_h_wmma(const float* __restrict__ y,
                   const float* __restrict__ ww,
                   const float* __restrict__ bw,
                   float* __restrict__ out) {
  __shared__ __align__(16) float ldsY[32 * HST];   // cols: w+127, w in [-127,256]
  __shared__ __align__(16) float wpad[WLEN];

  const int tid = threadIdx.x;
  const int nc  = blockIdx.x >> 2;
  const int c   = nc % DIMC;
  const int hq  = (blockIdx.x & 3) << 5;    // 0,32,64,96

  const float4 z4 = make_float4(0.f, 0.f, 0.f, 0.f);
  for (int i = tid; i < (32 * HST) / 4; i += 256) ((float4*)ldsY)[i] = z4;
  for (int i = tid; i < WLEN; i += 256) {
    int k = i - 16;
    wpad[i] = (k >= 0 && k < KTAP) ? ww[c * KTAP + k] : 0.0f;
  }
  __syncthreads();
  {
    const float* src = y + (size_t)nc * HW * HW + (size_t)hq * HW;
    int col = tid & 127, rb = tid >> 7;     // 128 lanes per row -> coalesced
    for (int i = 0; i < 16; ++i) {
      int r = rb + 2 * i;
      ldsY[r * HST + 127 + col] = src[r * HW + col];
    }
  }
  __syncthreads();

  const int lane = tid & 31;
  const int w0   = (tid >> 5) << 4;         // wave -> 0..112
  const int half = lane >> 4;
  const int ln   = lane & 15;
  const int kkb  = half << 1;

  const float bias = bw[c];
  v8f acc0, acc1;                           // h-tiles 0, 16
  for (int r = 0; r < 8; ++r) { acc0[r] = bias; acc1[r] = bias; }

  for (int j0 = 0; j0 < KPAD; j0 += 4) {
    int wi = 16 + j0 + kkb - ln;            // T[j0+kkb, n] = W[j0+kkb-n]
    v2f b = {wpad[wi], wpad[wi + 1]};
    int cc = w0 + j0 + kkb;                 // lds col = (w0-127+j) + 127
    const float* a0p = &ldsY[ln * HST + cc];
    const float* a1p = a0p + 16 * HST;
    v2f A0 = {a0p[0], a0p[1]};
    v2f A1 = {a1p[0], a1p[1]};
    acc0 = WMMA_F32X4(A0, b, acc0);
    acc1 = WMMA_F32X4(A1, b, acc1);
  }

  float* dst = out + (size_t)nc * HW * HW + (size_t)hq * HW;
  const int rb2 = half << 3;
  const int col = w0 + ln;
  for (int r = 0; r < 8; ++r) {
    dst[(rb2 + r) * HW + col]        = acc0[r];
    dst[(16 + rb2 + r) * HW + col]   = acc1[r];
  }
}

extern "C" void kernel_launch(void* const* d_in, const int* in_sizes, int n_in,
                              void* d_out, int out_size, void* d_ws, size_t ws_size,
                              hipStream_t stream) {
  (void)in_sizes; (void)n_in; (void)out_size; (void)ws_size;
  const float* x  = (const float*)d_in[0];
  const float* wh = (const float*)d_in[1];
  const float* bh = (const float*)d_in[2];
  const float* ww = (const float*)d_in[3];
  const float* bw = (const float*)d_in[4];
  float* out = (float*)d_out;
  float* mid = (float*)d_ws;   // 10*256*128*128 fp32 intermediate (168 MB)

  const int blocks = NB * DIMC * (HW / 32);   // 10240
  dwconv_v_wmma<<<blocks, 128, 0, stream>>>(x, wh, bh, mid);
  dwconv_h_wmma<<<blocks, 256, 0, stream>>>(mid, ww, bw, out);
}